// StarCTC_29609504538853
// MI455X (gfx1250) — compile-verified
//
#include <hip/hip_runtime.h>
#include <hip/hip_bf16.h>
#include <cstdint>

// ---------------- StarCTC constants (match reference) ----------------
#define VSIZE   4000
#define STAR_ID 2
#define PEN     0.0f
#define NEGV    (-1.0e30f)

#define MAXS    256      // >= 2*100 + 2 states; also == blockDim of forward kernel
#define MAXCOMP 128      // caps S at 256
#define MAXDEG  4        // max in-degree in the StarCTC lattice (by construction)

// per-batch graph block (ints): [S, fs0, fs1, pad] deg[MAXS] src[4*MAXS] lab[4*MAXS] wt[4*MAXS]
#define OFF_DEG 4
#define OFF_SRC (4 + MAXS)
#define OFF_LAB (4 + 5*MAXS)
#define OFF_WT  (4 + 9*MAXS)
#define GSTRIDE (4 + 13*MAXS)

typedef __attribute__((ext_vector_type(2))) float v2f;
typedef __attribute__((ext_vector_type(8))) float v8f;

// ---------------- Kernel A: per-utterance graph build (in-edge table by dst) ----------------
__global__ void starctc_build(const int* __restrict__ ys_pad,
                              const int* __restrict__ ylens,
                              int U, int* __restrict__ gws) {
  int b = blockIdx.x;
  if (threadIdx.x != 0) return;
  const int* ys = ys_pad + (size_t)b * U;
  int yl = ylens[b];

  int comp_l[MAXCOMP];
  unsigned char comp_w[MAXCOMP];
  int nc = 0;
  bool ws = false;
  auto feed = [&](int l) {
    if (l == STAR_ID) { ws = true; }
    else if (nc < MAXCOMP) { comp_l[nc] = l; comp_w[nc] = ws ? 1 : 0; nc++; ws = false; }
  };
  if (ys[0] != STAR_ID) feed(STAR_ID);
  for (int i = 0; i < yl; ++i) feed(ys[i]);
  if (ys[yl - 1] != STAR_ID) feed(STAR_ID);
  feed(-1);

  int* g    = gws + (size_t)b * GSTRIDE;
  int* deg  = g + OFF_DEG;
  int* isrc = g + OFF_SRC;
  int* ilab = g + OFF_LAB;
  int* iwt  = g + OFF_WT;
  for (int s = 0; s < MAXS; ++s) deg[s] = 0;

  int nf = 0;
  int fs[2] = {0, 0};
  auto addarc = [&](int s, int d, int l, float w) {
    if (l == -1) { if (nf < 2) fs[nf] = s; nf++; return; }   // final arc (weight 0)
    int j = deg[d];
    if (j < MAXDEG) {
      isrc[d * MAXDEG + j] = s;
      ilab[d * MAXDEG + j] = l;
      iwt [d * MAXDEG + j] = __float_as_int(w);
      deg[d] = j + 1;
    }
  };

  int label = comp_l[0];
  addarc(0, 0, 0, 0.f);
  if (comp_w[0]) addarc(0, 0, VSIZE, PEN);
  addarc(0, 1, label, 0.f);
  addarc(1, 1, label, 0.f);
  int idx = 1, prev = label;
  int n = nc;
  for (int count = 0; count < n - 1; ++count) {
    label = comp_l[count + 1];
    int wsx = comp_w[count + 1];
    if (label != prev) addarc(idx, idx + 2, label, 0.f);
    addarc(idx, idx + 1, 0, 0.f);
    if (wsx) addarc(idx, idx + 1, prev + VSIZE, PEN);
    addarc(idx + 1, idx + 1, 0, 0.f);
    if (wsx) addarc(idx + 1, idx + 1, VSIZE, PEN);
    addarc(idx + 1, idx + 2, label, 0.f);
    if (count < n - 2) addarc(idx + 2, idx + 2, label, 0.f);
    prev = label;
    idx += 2;
  }
  g[0] = idx + 1;   // #states (<= 256 by MAXCOMP cap)
  g[1] = fs[0];     // final-arc sources (fwt == 0)
  g[2] = fs[1];
}

// ---------------- Kernel B: nb[b,t] = logsumexp(nnet[b,t,1:V]) ----------------
// one wave per row; coalesced 512B/instr; cross-lane sum via V_WMMA_F32_16X16X4_F32
__global__ void starctc_rowlse(const float* __restrict__ nnet,
                               float* __restrict__ nb, int nrows) {
  int wid  = (int)((blockIdx.x * blockDim.x + threadIdx.x) >> 5);
  int lane = threadIdx.x & 31;
  if (wid >= nrows) return;                // wave-uniform: EXEC stays all-ones
  const float* row = nnet + (size_t)wid * VSIZE;

  float m = -INFINITY, s = 0.f;
  for (int c = lane * 4; c < VSIZE; c += 128) {
    __builtin_prefetch(row + c + 512, 0, 0);      // global_prefetch_b8
    float4 x = *(const float4*)(row + c);
    if (c == 0) x.x = -INFINITY;                  // exclude blank column 0
    float m4 = fmaxf(fmaxf(x.x, x.y), fmaxf(x.z, x.w));
    float s4 = __expf(x.x - m4) + __expf(x.y - m4) +
               __expf(x.z - m4) + __expf(x.w - m4);
    float mn = fmaxf(m, m4);
    s = s * __expf(m - mn) + s4 * __expf(m4 - mn);
    m = mn;
  }

  // wave-wide max (butterfly)
  float M = m;
#pragma unroll
  for (int off = 16; off >= 1; off >>= 1)
    M = fmaxf(M, __shfl_xor(M, off, 32));
  float sp = s * __expf(m - M);   // per-lane partial, rescaled to global max

  // D = A(16x4, a0=sp) * ones(4x16) => D[r,*] = sp_r + sp_{r+16}
  v2f a;  a.x = sp;  a.y = 0.f;
  v2f bo; bo.x = 1.f; bo.y = 1.f;
  v8f c = {};
  c = __builtin_amdgcn_wmma_f32_16x16x4_f32(false, a, false, bo,
                                            (short)0, c, false, false);
  float t = c[0] + c[1] + c[2] + c[3] + c[4] + c[5] + c[6] + c[7];
  t += __shfl_xor(t, 16, 32);     // combine low/high row groups -> total sum

  if (lane == 0) nb[wid] = fmaxf(M + __logf(t), NEGV);
}

// ---------------- Kernel C: forward recursion, one state per thread ----------------
// registers hold each state's <=4 in-arcs; 1 barrier/step; next-step gathers pipelined
__global__ void __launch_bounds__(MAXS)
starctc_forward(const float* __restrict__ nnet,
                const float* __restrict__ nb,
                const int* __restrict__ hlens,
                const int* __restrict__ gws,
                float* __restrict__ out, int T) {
  int b = blockIdx.x, tid = threadIdx.x;
  const int* g = gws + (size_t)b * GSTRIDE;
  int S = g[0], fs0 = g[1], fs1 = g[2];
  int hl = hlens[b];
  if (hl > T) hl = T;
  if (hl < 1) hl = 1;

  __shared__ float bufA[MAXS], bufB[MAXS];
  float* cur = bufA;
  float* nxt = bufB;

  // load this state's in-edges into registers
  int deg = 0, asrc[MAXDEG], acol[MAXDEG];
  bool isnb[MAXDEG], star[MAXDEG];
  float awt[MAXDEG];
  if (tid < S) deg = g[OFF_DEG + tid];
#pragma unroll
  for (int j = 0; j < MAXDEG; ++j) {
    asrc[j] = 0; acol[j] = 0; isnb[j] = false; star[j] = false; awt[j] = 0.f;
    if (j < deg) {
      asrc[j] = g[OFF_SRC + tid * MAXDEG + j];
      int l   = g[OFF_LAB + tid * MAXDEG + j];
      awt[j]  = __int_as_float(g[OFF_WT + tid * MAXDEG + j]);
      isnb[j] = (l == VSIZE);
      star[j] = (l > VSIZE);
      acol[j] = (l < VSIZE) ? l : ((l > VSIZE) ? (l - VSIZE) : 0);
    }
  }

  if (tid < MAXS) cur[tid] = (tid == 0) ? 0.f : NEGV;
  __syncthreads();

  const float* base = nnet + (size_t)b * T * VSIZE;

  // preload scores for t = 0
  float x[MAXDEG] = {0.f, 0.f, 0.f, 0.f};
  float nbv = nb[(size_t)b * T];
#pragma unroll
  for (int j = 0; j < MAXDEG; ++j)
    if (j < deg) x[j] = base[acol[j]];

  for (int t = 0; t < hl; ++t) {
    // issue next step's independent gathers now; latency hides behind compute+barrier
    float xn[MAXDEG] = {0.f, 0.f, 0.f, 0.f};
    float nbn = 0.f;
    if (t + 1 < hl) {
      const float* rown = base + (size_t)(t + 1) * VSIZE;
#pragma unroll
      for (int j = 0; j < MAXDEG; ++j)
        if (j < deg) xn[j] = rown[acol[j]];
      nbn = nb[(size_t)b * T + (t + 1)];
    }

    if (tid < S) {
      float vv[MAXDEG];
      float m = -INFINITY;
#pragma unroll
      for (int j = 0; j < MAXDEG; ++j) {
        if (j < deg) {
          float sc;
          if (isnb[j]) {
            sc = nbv;                                // non-blank LSE column
          } else if (star[j]) {                      // nbe: log(exp(nb) - exp(x))
            float xv = x[j];
            float m2 = fmaxf(nbv, xv);
            sc = m2 + __logf(__expf(nbv - m2) - __expf(xv - m2));
          } else {
            sc = x[j];
          }
          sc = fmaxf(sc, NEGV);
          float v = cur[asrc[j]] + awt[j] + sc;
          vv[j] = v;
          m = fmaxf(m, v);
        }
      }
      m = fmaxf(m, NEGV);
      float ssum = 0.f;
#pragma unroll
      for (int j = 0; j < MAXDEG; ++j)
        if (j < deg) ssum += __expf(vv[j] - m);
      nxt[tid] = m + __logf(ssum + 1e-37f);
    }
    __syncthreads();

    float* tmp = cur; cur = nxt; nxt = tmp;
#pragma unroll
    for (int j = 0; j < MAXDEG; ++j) x[j] = xn[j];
    nbv = nbn;
  }

  if (tid == 0) {
    float a0 = cur[fs0], a1 = cur[fs1];   // final-arc weights are 0
    float mm = fmaxf(a0, a1);
    out[b] = -(mm + __logf(__expf(a0 - mm) + __expf(a1 - mm)));
  }
}

// ---------------- launch ----------------
extern "C" void kernel_launch(void* const* d_in, const int* in_sizes, int n_in,
                              void* d_out, int out_size, void* d_ws, size_t ws_size,
                              hipStream_t stream) {
  (void)n_in; (void)out_size; (void)ws_size;
  const float* nnet  = (const float*)d_in[0];   // [B,T,V] fp32 log_softmax
  const int* ys_pad  = (const int*)d_in[1];     // [B,U]
  const int* hlens   = (const int*)d_in[2];     // [B]
  const int* ylens   = (const int*)d_in[3];     // [B]
  int B = in_sizes[2];
  int U = in_sizes[1] / B;
  int T = in_sizes[0] / (B * VSIZE);
  float* out = (float*)d_out;

  // workspace: nb[B*T] floats, then per-batch graph blocks
  float* nb = (float*)d_ws;
  size_t nb_bytes = ((size_t)B * T * sizeof(float) + 255) & ~(size_t)255;
  int* gws = (int*)((char*)d_ws + nb_bytes);

  starctc_build<<<B, 32, 0, stream>>>(ys_pad, ylens, U, gws);

  int nrows  = B * T;
  int blocks = (nrows + 7) / 8;   // 8 wave32s (256 threads) per block, 1 row/wave
  starctc_rowlse<<<blocks, 256, 0, stream>>>(nnet, nb, nrows);

  starctc_forward<<<B, MAXS, 0, stream>>>(nnet, nb, hlens, gws, out, T);
}